// MoEFeedForward_83537113907676
// MI455X (gfx1250) — compile-verified
//
#include <hip/hip_runtime.h>
#include <cstdint>

// MoE FFN, sparse top-2 + shared expert, fp32 WMMA (v_wmma_f32_16x16x4_f32)
// with CDNA5 async global<->LDS staging (GLOBAL_LOAD_ASYNC_TO_LDS_B128 /
// GLOBAL_STORE_ASYNC_FROM_LDS_B128, ASYNCcnt) and global_prefetch.
// T=4096 tokens, D=1024, H=2048, E=8, top-2.

#define T_TOK   4096
#define DM      1024
#define HID     2048
#define NE      8
#define ROWTILE 128
#define KCHUNK  32
#define MAXTILES 104   // <=72 routed row-tiles + 32 shared row-tiles
#define ASTRIDE 36     // 144 B rows: 16B aligned, bank-conflict-free
#define HSTRIDE 68     // 272 B rows: 16B aligned, bank-conflict-free

typedef float v2f __attribute__((ext_vector_type(2)));
typedef float v8f __attribute__((ext_vector_type(8)));

// ---- CDNA5 async global<->LDS helpers (inline asm; ASYNCcnt tracked) ----
__device__ __forceinline__ uint32_t ldsoff(const void* p) {
  // flat shared pointer: addr[31:0] is the workgroup-relative LDS byte offset
  return (uint32_t)(uintptr_t)p;
}
__device__ __forceinline__ void async_ld128(uint32_t lds, const float* gp) {
  asm volatile("global_load_async_to_lds_b128 %0, %1, off"
               :: "v"(lds), "v"((unsigned long long)(uintptr_t)gp)
               : "memory");
}
__device__ __forceinline__ void async_st128(float* gp, uint32_t lds) {
  asm volatile("global_store_async_from_lds_b128 %0, %1, off"
               :: "v"((unsigned long long)(uintptr_t)gp), "v"(lds)
               : "memory");
}
__device__ __forceinline__ void wait_async0() {
  asm volatile("s_wait_asynccnt 0x0" ::: "memory");
}

// ---------------- Router: wave per token ----------------
__global__ __launch_bounds__(256)
void k_router(const float* __restrict__ x, const float* __restrict__ wr,
              int* __restrict__ counts, int* __restrict__ ids,
              float* __restrict__ gates) {
  int wid  = threadIdx.x >> 5;
  int lane = threadIdx.x & 31;
  int t = blockIdx.x * 8 + wid;
  float acc[NE];
#pragma unroll
  for (int e = 0; e < NE; ++e) acc[e] = 0.f;
  const float* xr = x + (size_t)t * DM;
  for (int i = 0; i < DM / 32; ++i) {
    int d = lane + 32 * i;
    float xv = xr[d];
    const float4* w4 = (const float4*)(wr + (size_t)d * NE);
    float4 wa = w4[0], wb = w4[1];
    acc[0] += xv * wa.x; acc[1] += xv * wa.y;
    acc[2] += xv * wa.z; acc[3] += xv * wa.w;
    acc[4] += xv * wb.x; acc[5] += xv * wb.y;
    acc[6] += xv * wb.z; acc[7] += xv * wb.w;
  }
#pragma unroll
  for (int e = 0; e < NE; ++e) {
#pragma unroll
    for (int off = 16; off > 0; off >>= 1)
      acc[e] += __shfl_xor(acc[e], off, 32);
  }
  int i0 = 0; float l0 = acc[0];
#pragma unroll
  for (int e = 1; e < NE; ++e) if (acc[e] > l0) { l0 = acc[e]; i0 = e; }
  int i1 = (i0 == 0) ? 1 : 0; float l1 = acc[i1];
#pragma unroll
  for (int e = 0; e < NE; ++e)
    if (e != i0 && acc[e] > l1) { l1 = acc[e]; i1 = e; }
  // renormalized top-2 softmax: p0/(p0+p1) = 1/(1+exp(l1-l0))
  float g0 = 1.f / (1.f + __expf(l1 - l0));
  float g1 = 1.f - g0;
  if (lane == 0) {
    ids[t * 2]     = i0; ids[t * 2 + 1]   = i1;
    gates[t * 2]   = g0; gates[t * 2 + 1] = g1;
    atomicAdd(&counts[i0], 1);
    atomicAdd(&counts[i1], 1);
  }
}

// ------------- Scan: offsets + static tile table -------------
// ws ints: [0..7]=counts, [8..15]=cursor, [16]=ntiles, [32..]=table(int4)
__global__ void k_scan(int* __restrict__ ws_i) {
  if (threadIdx.x != 0 || blockIdx.x != 0) return;
  int* counts = ws_i;
  int* cursor = ws_i + 8;
  int4* table = (int4*)(ws_i + 32);
  int off = 0, t = 0;
  for (int e = 0; e < NE; ++e) {
    int c = counts[e];
    cursor[e] = off;
    int nt = (c + ROWTILE - 1) / ROWTILE;
    for (int i = 0; i < nt; ++i) {
      int valid = c - i * ROWTILE; if (valid > ROWTILE) valid = ROWTILE;
      table[t] = make_int4(e, off + i * ROWTILE, valid, t * ROWTILE);
      ++t;
    }
    off += c;
  }
  for (int i = 0; i < T_TOK / ROWTILE; ++i) {   // shared expert tiles
    table[t] = make_int4(NE, i * ROWTILE, ROWTILE, t * ROWTILE);
    ++t;
  }
  ws_i[16] = t;
}

// ------------- Scatter: per-expert token permutation -------------
__global__ void k_scatter(const int* __restrict__ ids,
                          const float* __restrict__ gates,
                          int* __restrict__ cursor,
                          int* __restrict__ ptok, float* __restrict__ pgate) {
  int t = blockIdx.x * blockDim.x + threadIdx.x;
#pragma unroll
  for (int k = 0; k < 2; ++k) {
    int e = ids[t * 2 + k];
    int pos = atomicAdd(&cursor[e], 1);
    ptok[pos]  = t;
    pgate[pos] = gates[t * 2 + k];
  }
}

// ------------- Pass A: g = silu(x@w1) * (x@w3), per tile -------------
// block: 256 thr (8 waves). Tile: 128 rows x 64 h-cols.
// wave -> (matrix m = wv>>2 in {w1,w3}, col-subtile c = wv&3). 8 C tiles/wave.
__global__ __launch_bounds__(256)
void k_up(const float* __restrict__ x,
          const float* __restrict__ w1, const float* __restrict__ w3,
          const float* __restrict__ sw1, const float* __restrict__ sw3,
          const int* __restrict__ ws_i, const int* __restrict__ ptok,
          float* __restrict__ g) {
  __shared__ float As[ROWTILE][ASTRIDE];
  __shared__ float Hs[ROWTILE][HSTRIDE];
  __shared__ int   toks[ROWTILE];

  int ntiles = ws_i[16];
  int tile = blockIdx.x;
  if (tile >= ntiles) return;
  int4 te = ((const int4*)(ws_i + 32))[tile];
  int e = te.x, seg = te.y, valid = te.z, gbase = te.w;
  int tid = threadIdx.x;
  if (tid < ROWTILE) {
    toks[tid] = (e < NE) ? ((tid < valid) ? ptok[seg + tid] : -1)
                         : (seg + tid);
  }
  __syncthreads();

  int wv = tid >> 5, lane = tid & 31;
  int m = wv >> 2, c = wv & 3;
  const float* W = (e < NE) ? ((m == 0 ? w1 : w3) + (size_t)e * DM * HID)
                            : (m == 0 ? sw1 : sw3);
  int h0   = blockIdx.y * 64;
  int hcol = h0 + c * 16 + (lane & 15);
  int kb   = (lane >> 4) * 2;         // K-pair owned by this half-wave

  v8f acc[8];
#pragma unroll
  for (int r = 0; r < 8; ++r)
#pragma unroll
    for (int v = 0; v < 8; ++v) acc[r][v] = 0.f;

  for (int kc = 0; kc < DM / KCHUNK; ++kc) {
    __syncthreads();   // all waves done reading As of previous chunk
    // async global->LDS stage of the (gathered) x tile, 16B granules
#pragma unroll
    for (int j = 0; j < 4; ++j) {
      int idx = tid + j * 256;
      int r = idx >> 3, q = idx & 7;
      int tok = toks[r];
      if (tok >= 0)
        async_ld128(ldsoff(&As[r][q * 4]),
                    x + (size_t)tok * DM + kc * KCHUNK + q * 4);
    }
    // prefetch next K-chunk of this wave's weight stream while async runs
    int kcn = (kc + 1 < DM / KCHUNK) ? kc + 1 : kc;
#pragma unroll
    for (int kk = 0; kk < KCHUNK / 4; ++kk)
      __builtin_prefetch(W + (size_t)(kcn * KCHUNK + kk * 4 + kb) * HID + hcol, 0, 1);
    wait_async0();
    __syncthreads();
#pragma unroll
    for (int kk = 0; kk < KCHUNK / 4; ++kk) {
      int kabs = kc * KCHUNK + kk * 4 + kb;
      v2f b;
      b.x = W[(size_t)kabs * HID + hcol];
      b.y = W[(size_t)(kabs + 1) * HID + hcol];
#pragma unroll
      for (int r = 0; r < 8; ++r) {
        int row = r * 16 + (lane & 15);
        v2f a;
        a.x = As[row][kk * 4 + kb];
        a.y = As[row][kk * 4 + kb + 1];
        acc[r] = __builtin_amdgcn_wmma_f32_16x16x4_f32(
            false, a, false, b, (short)0, acc[r], false, false);
      }
    }
  }
  __syncthreads();
  if (m == 0) {   // h1 waves publish to LDS
#pragma unroll
    for (int r = 0; r < 8; ++r)
#pragma unroll
      for (int v = 0; v < 8; ++v)
        Hs[r * 16 + (lane >> 4) * 8 + v][c * 16 + (lane & 15)] = acc[r][v];
  }
  __syncthreads();
  if (m == 1) {   // h3 waves fuse SwiGLU in place (same-thread RAW only)
#pragma unroll
    for (int r = 0; r < 8; ++r)
#pragma unroll
      for (int v = 0; v < 8; ++v) {
        int row = r * 16 + (lane >> 4) * 8 + v;
        int col = c * 16 + (lane & 15);
        float h1 = Hs[row][col];
        Hs[row][col] = (h1 / (1.f + __expf(-h1))) * acc[r][v];
      }
  }
  __syncthreads();
  // async LDS->global store of the g tile (completion at S_ENDPGM wait-idle)
#pragma unroll
  for (int j = 0; j < 8; ++j) {
    int idx = tid + j * 256;
    int r = idx >> 4, q = idx & 15;
    async_st128(g + (size_t)(gbase + r) * HID + h0 + q * 4,
                ldsoff(&Hs[r][q * 4]));
  }
}

// ------------- Pass B: out[tok] += gate * (g @ w2) -------------
// block: 256 thr. Tile: 128 rows x 64 d-cols. wave -> (row-half, col-subtile).
__global__ __launch_bounds__(256)
void k_down(const float* __restrict__ g,
            const float* __restrict__ w2, const float* __restrict__ sw2,
            const int* __restrict__ ws_i, const int* __restrict__ ptok,
            const float* __restrict__ pgate,
            float* __restrict__ out) {
  __shared__ float As[ROWTILE][ASTRIDE];
  __shared__ int   toks[ROWTILE];
  __shared__ float gts[ROWTILE];

  int ntiles = ws_i[16];
  int tile = blockIdx.x;
  if (tile >= ntiles) return;
  int4 te = ((const int4*)(ws_i + 32))[tile];
  int e = te.x, seg = te.y, valid = te.z, gbase = te.w;
  int tid = threadIdx.x;
  if (tid < ROWTILE) {
    if (e < NE) {
      toks[tid] = (tid < valid) ? ptok[seg + tid] : -1;
      gts[tid]  = (tid < valid) ? pgate[seg + tid] : 0.f;
    } else {
      toks[tid] = seg + tid;
      gts[tid]  = 1.f;
    }
  }
  __syncthreads();

  int wv = tid >> 5, lane = tid & 31;
  int rh = wv >> 2, c = wv & 3;
  const float* W = (e < NE) ? (w2 + (size_t)e * HID * DM) : sw2;
  int d0   = blockIdx.y * 64;
  int dcol = d0 + c * 16 + (lane & 15);
  int kb   = (lane >> 4) * 2;

  v8f acc[4];
#pragma unroll
  for (int r = 0; r < 4; ++r)
#pragma unroll
    for (int v = 0; v < 8; ++v) acc[r][v] = 0.f;

  for (int kc = 0; kc < HID / KCHUNK; ++kc) {
    __syncthreads();
    // async global->LDS stage of the g tile (contiguous rows)
#pragma unroll
    for (int j = 0; j < 4; ++j) {
      int idx = tid + j * 256;
      int r = idx >> 3, q = idx & 7;
      async_ld128(ldsoff(&As[r][q * 4]),
                  g + (size_t)(gbase + r) * HID + kc * KCHUNK + q * 4);
    }
    int kcn = (kc + 1 < HID / KCHUNK) ? kc + 1 : kc;
#pragma unroll
    for (int kk = 0; kk < KCHUNK / 4; ++kk)
      __builtin_prefetch(W + (size_t)(kcn * KCHUNK + kk * 4 + kb) * DM + dcol, 0, 1);
    wait_async0();
    __syncthreads();
#pragma unroll
    for (int kk = 0; kk < KCHUNK / 4; ++kk) {
      int kabs = kc * KCHUNK + kk * 4 + kb;
      v2f b;
      b.x = W[(size_t)kabs * DM + dcol];
      b.y = W[(size_t)(kabs + 1) * DM + dcol];
#pragma unroll
      for (int r = 0; r < 4; ++r) {
        int row = rh * 64 + r * 16 + (lane & 15);
        v2f a;
        a.x = As[row][kk * 4 + kb];
        a.y = As[row][kk * 4 + kb + 1];
        acc[r] = __builtin_amdgcn_wmma_f32_16x16x4_f32(
            false, a, false, b, (short)0, acc[r], false, false);
      }
    }
  }
#pragma unroll
  for (int r = 0; r < 4; ++r)
#pragma unroll
    for (int v = 0; v < 8; ++v) {
      int row = rh * 64 + r * 16 + (lane >> 4) * 8 + v;
      int tok = toks[row];
      if (tok >= 0)
        atomicAdd(&out[(size_t)tok * DM + dcol], gts[row] * acc[r][v]);
    }
}

// ---------------- Host launcher ----------------
extern "C" void kernel_launch(void* const* d_in, const int* in_sizes, int n_in,
                              void* d_out, int out_size, void* d_ws, size_t ws_size,
                              hipStream_t stream) {
  (void)in_sizes; (void)n_in; (void)ws_size;
  const float* x   = (const float*)d_in[0];
  const float* wr  = (const float*)d_in[1];
  const float* w1  = (const float*)d_in[2];
  const float* w3  = (const float*)d_in[3];
  const float* w2  = (const float*)d_in[4];
  const float* sw1 = (const float*)d_in[5];
  const float* sw3 = (const float*)d_in[6];
  const float* sw2 = (const float*)d_in[7];
  float* out = (float*)d_out;

  char* ws    = (char*)d_ws;
  int*  ws_i  = (int*)ws;                       // counts/cursor/ntiles/table
  int*  ids   = (int*)(ws + 2048);              // T*2 ints
  float* gates= (float*)(ws + 2048 + 32768);    // T*2 floats
  int*  ptok  = (int*)(ws + 2048 + 65536);      // T*2 ints
  float* pgate= (float*)(ws + 2048 + 98304);    // T*2 floats
  float* g    = (float*)(ws + (1u << 20));      // 13312 x 2048 f32 (~109 MB)

  hipMemsetAsync(ws, 0, 2048, stream);                              // counters
  hipMemsetAsync(d_out, 0, (size_t)out_size * sizeof(float), stream);

  k_router<<<T_TOK / 8, 256, 0, stream>>>(x, wr, ws_i, ids, gates);
  k_scan<<<1, 32, 0, stream>>>(ws_i);
  k_scatter<<<T_TOK / 256, 256, 0, stream>>>(ids, gates, ws_i + 8, ptok, pgate);

  dim3 gA(MAXTILES, HID / 64);
  k_up<<<gA, 256, 0, stream>>>(x, w1, w3, sw1, sw3, ws_i, ptok, g);

  dim3 gB(MAXTILES, DM / 64);
  k_down<<<gB, 256, 0, stream>>>(g, w2, sw2, ws_i, ptok, pgate, out);
}